// CrossAttention_38963943309809
// MI455X (gfx1250) — compile-verified
//
#include <hip/hip_runtime.h>
#include <math.h>

typedef __attribute__((ext_vector_type(2))) float v2f;
typedef __attribute__((ext_vector_type(8))) float v8f;

#define DIMX 1024
#define HDIM 512
#define NKEYS 65536

// workspace layout (float offsets)
#define WS_WVEC 0          // 1024
#define WS_QN   1024       // 1024
#define WS_QBUF 2048       // 512
#define WS_SCAL 2560       // 16: [0]=c [1]=sum_s [2]=sum_s2 [3]=mean [4]=alpha [5]=sum_e
#define WS_ACC  2576       // 1024
#define WS_U0   3600       // 1024
#define WS_U1   4624       // 1024
#define WS_OUTL 5648       // 1024
#define WS_SRAW 6672       // 65536
#define WS_INVN (6672 + 65536) // 65536

// ---------------- K0: init accumulators ----------------
__global__ void ca_init(float* ws, const float* bq, const float* bv, const float* bm) {
    int i = blockIdx.x * blockDim.x + threadIdx.x;   // 0..1023
    if (i < 512) ws[WS_QBUF + i] = bq[i];
    ws[WS_ACC + i]  = 0.0f;
    ws[WS_U1 + i]   = bv[i];
    ws[WS_OUTL + i] = bm[i];
    if (i < 16) ws[WS_SCAL + i] = 0.0f;
}

// ---------------- K1: qn = l2norm(q_init) ----------------
__global__ void ca_qnorm(const float* q_init, float* ws) {
    __shared__ float red[256];
    __shared__ float inv_s;
    int t = threadIdx.x;
    float s = 0.0f;
    for (int i = t; i < DIMX; i += 256) { float v = q_init[i]; s += v * v; }
    red[t] = s; __syncthreads();
    for (int off = 128; off > 0; off >>= 1) {
        if (t < off) red[t] += red[t + off];
        __syncthreads();
    }
    if (t == 0) inv_s = 1.0f / fmaxf(sqrtf(red[0]), 1e-12f);
    __syncthreads();
    float inv = inv_s;
    for (int i = t; i < DIMX; i += 256) ws[WS_QN + i] = q_init[i] * inv;
}

// ---------------- K2: q += qn @ Wq (8 blocks x 128 rows) ----------------
__global__ void ca_qproj(const float* Wq, float* ws) {
    int t = threadIdx.x;
    int base = blockIdx.x * 128;
    float a0 = 0.0f, a1 = 0.0f;
    for (int i = 0; i < 128; ++i) {
        float qv = ws[WS_QN + base + i];
        const float* wr = Wq + (size_t)(base + i) * HDIM;
        a0 = fmaf(qv, wr[t], a0);
        a1 = fmaf(qv, wr[t + 256], a1);
    }
    atomicAdd(&ws[WS_QBUF + t], a0);
    atomicAdd(&ws[WS_QBUF + t + 256], a1);
}

// ---------------- K3: wvec[i] = Wk_row_i . q ; block0 also c = q.bk ----------------
__global__ void ca_wvec(const float* Wk, const float* bk, float* ws) {
    int t = threadIdx.x;
    int row = blockIdx.x * 256 + t;
    const float4* wr = (const float4*)(Wk + (size_t)row * HDIM);
    const float4* qb = (const float4*)(ws + WS_QBUF);
    float d = 0.0f;
    for (int j = 0; j < HDIM / 4; ++j) {
        float4 w = wr[j]; float4 q = qb[j];
        d = fmaf(w.x, q.x, d); d = fmaf(w.y, q.y, d);
        d = fmaf(w.z, q.z, d); d = fmaf(w.w, q.w, d);
    }
    ws[WS_WVEC + row] = d;
    if (blockIdx.x == 0) {
        __shared__ float red[256];
        float p = ws[WS_QBUF + t] * bk[t] + ws[WS_QBUF + t + 256] * bk[t + 256];
        red[t] = p; __syncthreads();
        for (int off = 128; off > 0; off >>= 1) {
            if (t < off) red[t] += red[t + off];
            __syncthreads();
        }
        if (t == 0) ws[WS_SCAL + 0] = red[0];
    }
}

// ---------------- K4: WMMA streaming pass: raw scores + inv norms + stats ----------------
// Each wave handles 16 rows. A-tile: f32 16x4 per ISA layout; B broadcasts the wvec
// chunk across all 16 columns so every C column accumulates the row dots.
__global__ void ca_scores_wmma(const float* __restrict__ k_init, float* ws) {
    int lane = threadIdx.x & 31;
    int wave = blockIdx.x * (blockDim.x >> 5) + (threadIdx.x >> 5);
    int row0 = wave * 16;
    int m = lane & 15;           // row within tile
    int h = lane >> 4;           // K half: lanes 0-15 -> K%4 in {0,1}; 16-31 -> {2,3}
    const float* arow = k_init + (size_t)(row0 + m) * DIMX + 2 * h;
    const float* wp = ws + WS_WVEC + 2 * h;

    v8f c = {};
    float sq = 0.0f;
    for (int k = 0; k < DIMX; k += 4) {
        float2 a = *(const float2*)(arow + k);
        float2 b = *(const float2*)(wp + k);
        v2f av; av.x = a.x; av.y = a.y;
        v2f bv; bv.x = b.x; bv.y = b.y;
        c = __builtin_amdgcn_wmma_f32_16x16x4_f32(false, av, false, bv,
                                                  (short)0, c, false, false);
        sq = fmaf(a.x, a.x, sq);
        sq = fmaf(a.y, a.y, sq);
    }
    // full sum-of-squares for row m lands in lanes m and m+16
    sq += __shfl_xor(sq, 16, 32);

    // C column 0: lane 0 holds rows 0..7 in c[0..7]; lane 16 holds rows 8..15
    float dot = 0.0f;
#pragma unroll
    for (int r = 0; r < 8; ++r) {
        float d0 = __shfl(c[r], 0, 32);
        float d1 = __shfl(c[r], 16, 32);
        dot = (lane == r) ? d0 : dot;
        dot = (lane == r + 8) ? d1 : dot;
    }

    if (lane < 16) {
        float invn = 1.0f / fmaxf(sqrtf(sq), 1e-12f);
        float s = fmaf(dot, invn, ws[WS_SCAL + 0]);
        ws[WS_SRAW + row0 + lane] = s;
        ws[WS_INVN + row0 + lane] = invn;
        float ssum = s, s2sum = s * s;
        for (int off = 8; off >= 1; off >>= 1) {
            ssum  += __shfl_xor(ssum, off, 32);
            s2sum += __shfl_xor(s2sum, off, 32);
        }
        if (lane == 0) {
            atomicAdd(&ws[WS_SCAL + 1], ssum);
            atomicAdd(&ws[WS_SCAL + 2], s2sum);
        }
    }
}

// ---------------- K5: mean / unbiased std ----------------
__global__ void ca_stats(float* ws) {
    if (threadIdx.x == 0) {
        float n = (float)NKEYS;
        float sum_s = ws[WS_SCAL + 1], sum_s2 = ws[WS_SCAL + 2];
        float mean = sum_s / n;
        float var = fmaxf((sum_s2 - sum_s * mean) / (n - 1.0f), 0.0f);
        ws[WS_SCAL + 3] = mean;
        ws[WS_SCAL + 4] = 1.0f / (sqrtf(var) + 1e-8f);
    }
}

// ---------------- K6: second stream: Sum e and Sum e*invn*row ----------------
__global__ void ca_weighted(const float* __restrict__ k_init, float* ws) {
    int t = threadIdx.x;
    int n0 = blockIdx.x * 64;
    float mean = ws[WS_SCAL + 3];
    float alpha = ws[WS_SCAL + 4];
    float4 acc = make_float4(0.f, 0.f, 0.f, 0.f);
    float esum = 0.0f;
    for (int r = 0; r < 64; ++r) {
        int n = n0 + r;
        float s = ws[WS_SRAW + n];
        float invn = ws[WS_INVN + n];
        float z = fminf(fmaxf((s - mean) * alpha, -10.0f), 10.0f);
        float e = expf(z - 10.0f);
        float w = e * invn;
        const float* rowp = k_init + (size_t)n * DIMX + t * 4;
        float4 x = *(const float4*)rowp;
        __builtin_prefetch(rowp + DIMX, 0, 1);   // next row, same columns
        acc.x = fmaf(w, x.x, acc.x);
        acc.y = fmaf(w, x.y, acc.y);
        acc.z = fmaf(w, x.z, acc.z);
        acc.w = fmaf(w, x.w, acc.w);
        esum += (t == 0) ? e : 0.0f;
    }
    atomicAdd(&ws[WS_ACC + t * 4 + 0], acc.x);
    atomicAdd(&ws[WS_ACC + t * 4 + 1], acc.y);
    atomicAdd(&ws[WS_ACC + t * 4 + 2], acc.z);
    atomicAdd(&ws[WS_ACC + t * 4 + 3], acc.w);
    if (t == 0) atomicAdd(&ws[WS_SCAL + 5], esum);
}

// ---------------- K7: u0 = acc / sum_e ----------------
__global__ void ca_u0(float* ws) {
    int t = threadIdx.x;
    float invS = 1.0f / ws[WS_SCAL + 5];
    for (int i = t; i < DIMX; i += 256) ws[WS_U0 + i] = ws[WS_ACC + i] * invS;
}

// ---------------- K8/K9: target += u @ W (16 blocks x 64 rows) ----------------
__global__ void ca_proj1024(const float* __restrict__ W, float* ws, int u_off, int t_off) {
    int t = threadIdx.x;
    int i0 = blockIdx.x * 64;
    float a0 = 0.f, a1 = 0.f, a2 = 0.f, a3 = 0.f;
    for (int i = 0; i < 64; ++i) {
        float uv = ws[u_off + i0 + i];
        const float* wr = W + (size_t)(i0 + i) * DIMX;
        a0 = fmaf(uv, wr[t], a0);
        a1 = fmaf(uv, wr[t + 256], a1);
        a2 = fmaf(uv, wr[t + 512], a2);
        a3 = fmaf(uv, wr[t + 768], a3);
    }
    atomicAdd(&ws[t_off + t], a0);
    atomicAdd(&ws[t_off + t + 256], a1);
    atomicAdd(&ws[t_off + t + 512], a2);
    atomicAdd(&ws[t_off + t + 768], a3);
}

// ---------------- K10: gated residual ----------------
__global__ void ca_final(const float* q_init, const float* gamma, const float* ws, float* out) {
    int t = threadIdx.x;
    float g = 1.0f / (1.0f + expf(-gamma[0]));
    for (int i = t; i < DIMX; i += 256)
        out[i] = g * q_init[i] + (1.0f - g) * ws[WS_OUTL + i];
}

extern "C" void kernel_launch(void* const* d_in, const int* in_sizes, int n_in,
                              void* d_out, int out_size, void* d_ws, size_t ws_size,
                              hipStream_t stream) {
    const float* q_init = (const float*)d_in[0];
    const float* k_init = (const float*)d_in[1];
    const float* Wq = (const float*)d_in[2];
    const float* bq = (const float*)d_in[3];
    const float* Wk = (const float*)d_in[4];
    const float* bk = (const float*)d_in[5];
    const float* Wv = (const float*)d_in[6];
    const float* bv = (const float*)d_in[7];
    const float* Wm = (const float*)d_in[8];
    const float* bm = (const float*)d_in[9];
    const float* gamma = (const float*)d_in[10];
    float* ws = (float*)d_ws;
    float* out = (float*)d_out;

    ca_init<<<4, 256, 0, stream>>>(ws, bq, bv, bm);
    ca_qnorm<<<1, 256, 0, stream>>>(q_init, ws);
    ca_qproj<<<8, 256, 0, stream>>>(Wq, ws);
    ca_wvec<<<4, 256, 0, stream>>>(Wk, bk, ws);
    ca_scores_wmma<<<512, 256, 0, stream>>>(k_init, ws);   // 512*8 waves * 16 rows = 65536
    ca_stats<<<1, 32, 0, stream>>>(ws);
    ca_weighted<<<1024, 256, 0, stream>>>(k_init, ws);     // 1024*64 rows = 65536
    ca_u0<<<1, 256, 0, stream>>>(ws);
    ca_proj1024<<<16, 256, 0, stream>>>(Wv, ws, WS_U0, WS_U1);
    ca_proj1024<<<16, 256, 0, stream>>>(Wm, ws, WS_U1, WS_OUTL);
    ca_final<<<1, 256, 0, stream>>>(q_init, gamma, ws, out);
}